// SparseLayer_56925496541735
// MI455X (gfx1250) — compile-verified
//
#include <hip/hip_runtime.h>
#include <stdint.h>
#include <stddef.h>

#define CCHUNK 16      // nnz per async chunk (rows padded to multiples of this)
#define MBLOCK 128     // threads per block in main kernel (4 waves, 512 bt via float4)
#define RPB    16      // rows per block in main kernel

// ---------------- JAX threefry2x32 (exact) ----------------
__device__ __forceinline__ void tf_round(uint32_t& x0, uint32_t& x1, int r) {
  x0 += x1; x1 = (x1 << r) | (x1 >> (32 - r)); x1 ^= x0;
}
__device__ __forceinline__ void threefry2x32(uint32_t k0, uint32_t k1,
                                             uint32_t& x0, uint32_t& x1) {
  uint32_t ks2 = k0 ^ k1 ^ 0x1BD11BDAu;
  x0 += k0; x1 += k1;
  tf_round(x0,x1,13); tf_round(x0,x1,15); tf_round(x0,x1,26); tf_round(x0,x1,6);
  x0 += k1;  x1 += ks2 + 1u;
  tf_round(x0,x1,17); tf_round(x0,x1,29); tf_round(x0,x1,16); tf_round(x0,x1,24);
  x0 += ks2; x1 += k0 + 2u;
  tf_round(x0,x1,13); tf_round(x0,x1,15); tf_round(x0,x1,26); tf_round(x0,x1,6);
  x0 += k0;  x1 += k1 + 3u;
  tf_round(x0,x1,17); tf_round(x0,x1,29); tf_round(x0,x1,16); tf_round(x0,x1,24);
  x0 += k1;  x1 += ks2 + 4u;
  tf_round(x0,x1,13); tf_round(x0,x1,15); tf_round(x0,x1,26); tf_round(x0,x1,6);
  x0 += ks2; x1 += k0 + 5u;
}
// Count of 10 draws (uniform<0.1) for one (b,t); key = fold_in(key(42), 1)
__device__ __forceinline__ float bg_count(int bt, int BT) {
  uint32_t f0 = 0u, f1 = 1u;
  threefry2x32(0u, 42u, f0, f1);           // folded key
  int n = BT * 10, half = n / 2;
  int cnt = 0;
  for (int d = 0; d < 10; ++d) {
    int idx = bt * 10 + d;
    int p = idx < half ? idx : idx - half;
    uint32_t x0 = (uint32_t)p, x1 = (uint32_t)(p + half);
    threefry2x32(f0, f1, x0, x1);
    uint32_t bits = idx < half ? x0 : x1;
    float u = __uint_as_float((bits >> 9) | 0x3f800000u) - 1.0f;
    u = fmaxf(u, 0.0f);
    if (u < 0.1f) cnt++;
  }
  return (float)cnt;
}

// ---------------- prep kernels ----------------
__global__ void zero_i32_kernel(int* p, int n) {
  int i = blockIdx.x * blockDim.x + threadIdx.x;
  if (i < n) p[i] = 0;
}

__global__ void noise_kernel(float* R, int BT) {
  int bt = blockIdx.x * blockDim.x + threadIdx.x;
  if (bt < BT) R[bt] = bg_count(bt, BT);
}

__global__ void hist_kernel(const int* __restrict__ indices, int* cnt, int nnz) {
  int i = blockIdx.x * blockDim.x + threadIdx.x;
  if (i < nnz) atomicAdd(&cnt[indices[2 * i]], 1);
}

// exclusive scan of padded row lengths; single 1024-thread block
__global__ void scan_kernel(const int* __restrict__ cnt, int* ptr, int n) {
  __shared__ int sh[1024];
  __shared__ int carry_s;
  int tid = threadIdx.x;
  if (tid == 0) carry_s = 0;
  __syncthreads();
  for (int base = 0; base < n; base += 1024) {
    int i = base + tid;
    int v = (i < n) ? ((cnt[i] + CCHUNK - 1) / CCHUNK) * CCHUNK : 0;
    sh[tid] = v;
    __syncthreads();
    for (int off = 1; off < 1024; off <<= 1) {
      int t = (tid >= off) ? sh[tid - off] : 0;
      __syncthreads();
      sh[tid] += t;
      __syncthreads();
    }
    int incl = sh[tid];
    int carry = carry_s;
    if (i < n) ptr[i] = carry + incl - v;
    __syncthreads();
    if (tid == 1023) carry_s = carry + incl;
    __syncthreads();
  }
  if (tid == 0) ptr[n] = carry_s;
}

__global__ void scatter_kernel(const int* __restrict__ indices,
                               const int* __restrict__ ptr, int* fill, int* org, int nnz) {
  int i = blockIdx.x * blockDim.x + threadIdx.x;
  if (i >= nnz) return;
  int r = indices[2 * i];
  int pos = ptr[r] + atomicAdd(&fill[r], 1);
  org[pos] = i;
}

// per-row: sort original indices (deterministic FP order), emit padded col/w
__global__ void rowsort_kernel(const int* __restrict__ indices, const float* __restrict__ wts,
                               const int* __restrict__ ptr, const int* __restrict__ cnt,
                               int* org, int* col, float* w, int nrows) {
  int r = blockIdx.x * blockDim.x + threadIdx.x;
  if (r >= nrows) return;
  int s = ptr[r], c = cnt[r], e2 = ptr[r + 1];
  for (int i = 1; i < c; ++i) {          // insertion sort (avg ~40 elems)
    int key = org[s + i];
    int j = i - 1;
    while (j >= 0 && org[s + j] > key) { org[s + j + 1] = org[s + j]; --j; }
    org[s + j + 1] = key;
  }
  for (int k = 0; k < c; ++k) {
    int oi = org[s + k];
    col[s + k] = indices[2 * oi + 1];
    w[s + k] = wts[oi];
  }
  for (int k = s + c; k < e2; ++k) { col[k] = 0; w[k] = 0.0f; }
}

__global__ void transpose_kernel(const float* __restrict__ x, float* __restrict__ xT,
                                 int BT, int NIN) {
  __shared__ float t[32][33];
  int c0 = blockIdx.x * 32, b0 = blockIdx.y * 32;
  int tx = threadIdx.x, ty = threadIdx.y;  // 32x8
  for (int j = 0; j < 32; j += 8) {
    int bt = b0 + ty + j, c = c0 + tx;
    if (bt < BT && c < NIN) t[ty + j][tx] = x[(size_t)bt * NIN + c];
  }
  __syncthreads();
  for (int j = 0; j < 32; j += 8) {
    int c = c0 + ty + j, bt = b0 + tx;
    if (bt < BT && c < NIN) xT[(size_t)c * BT + bt] = t[tx][ty + j];
  }
}

// ------- main SpMM: double-buffered async-LDS gather pipeline + FMA -------
__global__ __launch_bounds__(MBLOCK) void spmm_async_kernel(
    const int* __restrict__ col, const float* __restrict__ w,
    const int* __restrict__ ptr, const float* __restrict__ xT,
    const float* __restrict__ R, const float* __restrict__ bkg,
    float* __restrict__ out, int BT, int nrows) {
  __shared__ __align__(16) float4 stage[2][CCHUNK][MBLOCK];  // 64 KB, lane-private slots
  const int tid = threadIdx.x;
  const int r0 = blockIdx.x * RPB;
  const int bt = blockIdx.y * (MBLOCK * 4) + tid * 4;
  if (bt >= BT || r0 >= nrows) return;

  const float4 Rv = *(const float4*)(R + bt);
  // low 32 bits of a flat shared-aperture pointer == LDS byte offset
  const uint32_t lds_lane = (uint32_t)(uintptr_t)(&stage[0][0][0]) + (uint32_t)tid * 16u;
  const float* xrow = xT + bt;
  const int r_end = (r0 + RPB < nrows) ? r0 + RPB : nrows;

  auto noise_acc = [&](int r) {
    float bw = bkg[r];
    float4 a;
    a.x = (bw * Rv.x) / 10.0f;
    a.y = (bw * Rv.y) / 10.0f;
    a.z = (bw * Rv.z) / 10.0f;
    a.w = (bw * Rv.w) / 10.0f;
    return a;
  };
  auto store_row = [&](int r, const float4& a) {
    float* o = out + (size_t)bt * nrows + r;   // out[bt][r]
    o[0] = a.x;
    o[(size_t)nrows] = a.y;
    o[2 * (size_t)nrows] = a.z;
    o[3 * (size_t)nrows] = a.w;
  };
  auto issue_chunk = [&](int k0, int par) {
#pragma unroll
    for (int j = 0; j < CCHUNK; ++j) {
      int c = col[k0 + j];
      uint64_t ga = (uint64_t)(uintptr_t)(xrow + (size_t)c * BT);
      uint32_t la = lds_lane + (uint32_t)(par * (CCHUNK * MBLOCK * 16))
                             + (uint32_t)(j * (MBLOCK * 16));
      asm volatile("global_load_async_to_lds_b128 %0, %1, off"
                   :: "v"(la), "v"(ga) : "memory");
    }
  };

  const int sAll = ptr[r0];
  const int eAll = ptr[r_end];

  // consumer-side row tracking; flush noise-only (empty) rows up front
  int rcur = r0;
  while (rcur < r_end && ptr[rcur + 1] == ptr[rcur]) {
    store_row(rcur, noise_acc(rcur));
    ++rcur;
  }
  float4 acc = {0.f, 0.f, 0.f, 0.f};
  int ecur = 0;
  if (rcur < r_end) { acc = noise_acc(rcur); ecur = ptr[rcur + 1]; }

  const int nch = (eAll - sAll) / CCHUNK;
  if (nch > 0) issue_chunk(sAll, 0);            // prologue

  for (int ci = 0; ci < nch; ++ci) {
    const int k0 = sAll + ci * CCHUNK;
    const bool has_next = (ci + 1 < nch);
    if (has_next) {
      // WAR guard: LDS reads of chunk ci-1 (same buffer) must retire before rewrite
      asm volatile("s_wait_dscnt 0x0" ::: "memory");
      issue_chunk(k0 + CCHUNK, (ci + 1) & 1);
      // <=16 outstanding ==> chunk ci has fully landed in LDS
      asm volatile("s_wait_asynccnt 0x10" ::: "memory");
    } else {
      asm volatile("s_wait_asynccnt 0x0" ::: "memory");
    }
    const int par = ci & 1;
#pragma unroll
    for (int j = 0; j < CCHUNK; ++j) {
      float wj = w[k0 + j];
      float4 xv = stage[par][j][tid];
      acc.x += wj * xv.x; acc.y += wj * xv.y;
      acc.z += wj * xv.z; acc.w += wj * xv.w;
    }
    if (k0 + CCHUNK == ecur) {                  // row boundary (chunk-aligned)
      store_row(rcur, acc);
      ++rcur;
      while (rcur < r_end && ptr[rcur + 1] == ptr[rcur]) {
        store_row(rcur, noise_acc(rcur));
        ++rcur;
      }
      if (rcur < r_end) { acc = noise_acc(rcur); ecur = ptr[rcur + 1]; }
    }
  }
}

// ---------------- fallback (ws too small): init + atomic scatter ----------------
__global__ void fb_init_kernel(const float* __restrict__ bkg, float* __restrict__ out,
                               int BT, int nrows) {
  size_t i = (size_t)blockIdx.x * blockDim.x + threadIdx.x;
  size_t total = (size_t)BT * nrows;
  if (i >= total) return;
  int bt = (int)(i / nrows), r = (int)(i % nrows);
  out[i] = (bkg[r] * bg_count(bt, BT)) / 10.0f;
}
__global__ void fb_scatter_kernel(const int* __restrict__ indices,
                                  const float* __restrict__ wts,
                                  const float* __restrict__ x,
                                  float* __restrict__ out, int BT, int NIN, int nrows) {
  int i = blockIdx.x;              // one nnz per block
  int r = indices[2 * i], c = indices[2 * i + 1];
  float wv = wts[i];
  for (int bt = threadIdx.x; bt < BT; bt += blockDim.x)
    atomicAdd(&out[(size_t)bt * nrows + r], wv * x[(size_t)bt * NIN + c]);
}

// ---------------- launch ----------------
extern "C" void kernel_launch(void* const* d_in, const int* in_sizes, int n_in,
                              void* d_out, int out_size, void* d_ws, size_t ws_size,
                              hipStream_t stream) {
  const float* x   = (const float*)d_in[0];
  const int* idx   = (const int*)d_in[1];
  const float* wts = (const float*)d_in[2];
  const float* bkg = (const float*)d_in[3];
  float* out = (float*)d_out;

  const int NNZ   = in_sizes[2];
  const int NROWS = in_sizes[3];
  const int BT    = out_size / NROWS;
  const int NIN   = in_sizes[0] / BT;

  // workspace layout (256B aligned)
  size_t off = 0;
  auto take = [&](size_t bytes) { size_t o = off; off += (bytes + 255) & ~(size_t)255; return o; };
  const size_t padmax = (size_t)NNZ + (size_t)NROWS * (CCHUNK - 1);
  size_t o_R    = take((size_t)BT * 4);
  size_t o_cnt  = take((size_t)NROWS * 4);
  size_t o_fill = take((size_t)NROWS * 4);
  size_t o_ptr  = take(((size_t)NROWS + 1) * 4);
  size_t o_org  = take(padmax * 4);
  size_t o_col  = take(padmax * 4);
  size_t o_w    = take(padmax * 4);
  size_t o_xT   = take((size_t)NIN * BT * 4);
  const size_t needed = off;

  if (ws_size < needed) {  // fallback: no-workspace atomic path
    size_t total = (size_t)BT * NROWS;
    fb_init_kernel<<<dim3((unsigned)((total + 255) / 256)), 256, 0, stream>>>(bkg, out, BT, NROWS);
    fb_scatter_kernel<<<dim3((unsigned)NNZ), 256, 0, stream>>>(idx, wts, x, out, BT, NIN, NROWS);
    return;
  }

  char* ws = (char*)d_ws;
  float* R   = (float*)(ws + o_R);
  int* cnt   = (int*)(ws + o_cnt);
  int* fill  = (int*)(ws + o_fill);
  int* ptr   = (int*)(ws + o_ptr);
  int* org   = (int*)(ws + o_org);
  int* col   = (int*)(ws + o_col);
  float* w   = (float*)(ws + o_w);
  float* xT  = (float*)(ws + o_xT);

  zero_i32_kernel<<<dim3((NROWS + 255) / 256), 256, 0, stream>>>(cnt, NROWS);
  zero_i32_kernel<<<dim3((NROWS + 255) / 256), 256, 0, stream>>>(fill, NROWS);
  noise_kernel<<<dim3((BT + 255) / 256), 256, 0, stream>>>(R, BT);
  hist_kernel<<<dim3((NNZ + 255) / 256), 256, 0, stream>>>(idx, cnt, NNZ);
  scan_kernel<<<1, 1024, 0, stream>>>(cnt, ptr, NROWS);
  scatter_kernel<<<dim3((NNZ + 255) / 256), 256, 0, stream>>>(idx, ptr, fill, org, NNZ);
  rowsort_kernel<<<dim3((NROWS + 255) / 256), 256, 0, stream>>>(idx, wts, ptr, cnt, org, col, w, NROWS);
  transpose_kernel<<<dim3((NIN + 31) / 32, (BT + 31) / 32), dim3(32, 8), 0, stream>>>(x, xT, BT, NIN);
  spmm_async_kernel<<<dim3((NROWS + RPB - 1) / RPB, (BT + MBLOCK * 4 - 1) / (MBLOCK * 4)),
                     MBLOCK, 0, stream>>>(col, w, ptr, xT, R, bkg, out, BT, NROWS);
}